// FlowLossProb_61692910239837
// MI455X (gfx1250) — compile-verified
//
#include <hip/hip_runtime.h>
#include <hip/hip_bf16.h>

// ---------------------------------------------------------------------------
// FlowLossProb for MI455X (gfx1250, wave32).
// Memory-bound reduction: ~403 MB fp32 streamed once -> 1 scalar.
// Pass 1: fully-coalesced streaming compute + per-block LDS reduction.
// Pass 2: single-wave deterministic reduction using V_WMMA_F32_16X16X4_F32
//         (B = ones => D[m][n] = rowsum(A); 4 accumulating WMMAs consume all
//          256 per-lane partials), then shfl_xor(16) for the final 16->1.
// ---------------------------------------------------------------------------

#define Hh 512
#define Ww 512
#define Bb 8
#define HW (Hh * Ww)
#define NBLK 2048
#define NTHR 256

__device__ __forceinline__ float clipf(float v) {
    const float lo = 1e-10f;
    const float hi = 1.0f - 1e-10f;
    return fminf(fmaxf(v, lo), hi);
}

__global__ void __launch_bounds__(NTHR)
flowloss_main(const float* __restrict__ roi,
              const float* __restrict__ hm0,
              const float* __restrict__ hm1,
              const float* __restrict__ hm2,
              const float* __restrict__ r1pf,
              const float* __restrict__ r1pb,
              const float* __restrict__ r1nf,
              const float* __restrict__ r1nb,
              const float* __restrict__ r0f,
              const float* __restrict__ r0b,
              const float* __restrict__ r2f,
              const float* __restrict__ r2b,
              const float* __restrict__ f01f,   // flow pair 1: 'a'
              const float* __restrict__ f10b,   // flow pair 1: 'b' (shifted)
              const float* __restrict__ f12f,   // flow pair 2: 'a'
              const float* __restrict__ f21b,   // flow pair 2: 'b' (shifted)
              float* __restrict__ partials) {
    const int stride = gridDim.x * blockDim.x;           // 524288
    const int total  = Bb * HW;                          // 2097152
    float acc = 0.0f;

    for (int p = blockIdx.x * blockDim.x + threadIdx.x; p < total; p += stride) {
        const int x = p & (Ww - 1);
        const int y = (p >> 9) & (Hh - 1);
        const int b = p >> 18;

        // ---- reconstruction losses (all streams coalesced, index == p) ----
        const float m  = roi[p];
        const float h0 = hm0[p];
        const float h1 = hm1[p];
        const float h2 = hm2[p];
        const float pre  = fabsf(h0 - h1);
        const float post = fabsf(h1 - h2);
        const float wp = m * (1.0f + pre);    // RF = 1.0
        const float wq = m * (1.0f + post);

        {
            float d;
            float s1p = 0.0f, s1n = 0.0f, s0 = 0.0f, s2 = 0.0f;
            d = r1pf[p] - h1; s1p += d * d;
            d = r1pb[p] - h1; s1p += d * d;
            d = r1nf[p] - h1; s1n += d * d;
            d = r1nb[p] - h1; s1n += d * d;
            d = r0f[p]  - h0; s0  += d * d;
            d = r0b[p]  - h0; s0  += d * d;
            d = r2f[p]  - h2; s2  += d * d;
            d = r2b[p]  - h2; s2  += d * d;
            acc += 0.25f * (s1p * wp + s1n * wq);
            acc += 0.5f  * (s0 * wp + s2 * wq);
        }

        // ---- flow consistency -------------------------------------------
        // channel i: dy = {1,1,1, 0,0,0, -1,-1,-1}[i], dx = {1,0,-1}[i%3]
        // a = flow[i](y,x); b = flow_inv[8-i](y-dy, x-dx); mask = roi(y,x)
        const int cbase = (b * 9 * Hh + y) * Ww + x;     // channel-0 element
        float cacc = 0.0f;
#pragma unroll
        for (int i = 0; i < 9; ++i) {
            const int dy = (i < 3) ? 1 : (i < 6) ? 0 : -1;
            const int mm = i % 3;
            const int dx = (mm == 0) ? 1 : (mm == 1) ? 0 : -1;
            const int yb = y - dy;
            const int xb = x - dx;
            if ((unsigned)yb < (unsigned)Hh && (unsigned)xb < (unsigned)Ww) {
                const int aoff = cbase + i * HW;
                const int boff = cbase + (8 - i) * HW - dy * Ww - dx;
                const float t1 = clipf(f01f[aoff]) - clipf(f10b[boff]);
                const float t2 = clipf(f12f[aoff]) - clipf(f21b[boff]);
                cacc += (t1 * t1 + t2 * t2) * m;
            }
        }
        acc += cacc * (1.0f / 9.0f);
    }

    // ---- block reduction (deterministic tree) ---------------------------
    __shared__ float red[NTHR];
    red[threadIdx.x] = acc;
    __syncthreads();
#pragma unroll
    for (int s = NTHR / 2; s > 0; s >>= 1) {
        if (threadIdx.x < s) red[threadIdx.x] += red[threadIdx.x + s];
        __syncthreads();
    }
    if (threadIdx.x == 0) partials[blockIdx.x] = red[0];
}

// ---------------------------------------------------------------------------
// Final reduction: one wave32.  256 per-lane accumulators (8/lane) are fed
// through V_WMMA_F32_16X16X4_F32 with B = ones:  D[m][n] = sum_k A[m][k].
// Summing the 8 C/D VGPRs per lane gives column sums of rows 0-7 (lanes
// 0-15) and rows 8-15 (lanes 16-31); shfl_xor(16) completes the total.
// ---------------------------------------------------------------------------
typedef __attribute__((ext_vector_type(2))) float v2f;
typedef __attribute__((ext_vector_type(8))) float v8f;

__global__ void __launch_bounds__(32)
flowloss_reduce(const float* __restrict__ partials, int n, float* __restrict__ out) {
    const int lane = threadIdx.x;                        // 0..31, EXEC all-ones
    float acc[8] = {0.f, 0.f, 0.f, 0.f, 0.f, 0.f, 0.f, 0.f};
    for (int i = lane; i < n; i += 32) acc[(i >> 5) & 7] += partials[i];

    v8f c = {};
    v2f onesv;
    onesv[0] = 1.0f;
    onesv[1] = 1.0f;
#pragma unroll
    for (int ch = 0; ch < 4; ++ch) {
        v2f a;
        a[0] = acc[2 * ch];
        a[1] = acc[2 * ch + 1];
        // 8 args: (neg_a, A, neg_b, B, c_mod, C, reuse_a, reuse_b)
        c = __builtin_amdgcn_wmma_f32_16x16x4_f32(false, a, false, onesv,
                                                  (short)0, c, false, false);
    }
    float s = c[0] + c[1] + c[2] + c[3] + c[4] + c[5] + c[6] + c[7];
    s += __shfl_xor(s, 16, 32);
    if (lane == 0) out[0] = s;
}

// ---------------------------------------------------------------------------
extern "C" void kernel_launch(void* const* d_in, const int* in_sizes, int n_in,
                              void* d_out, int out_size, void* d_ws, size_t ws_size,
                              hipStream_t stream) {
    // input order (setup_inputs): roi_mask, boundary_mask(unused), hm_0, hm_1,
    // hm_2, rec_1pf, rec_1pb, rec_1nf, rec_1nb, rec_0f, rec_0b, rec_2f, rec_2b,
    // flow_0_1f, flow_1_0b, flow_1_2f, flow_2_1b
    const float* roi  = (const float*)d_in[0];
    const float* hm0  = (const float*)d_in[2];
    const float* hm1  = (const float*)d_in[3];
    const float* hm2  = (const float*)d_in[4];
    const float* r1pf = (const float*)d_in[5];
    const float* r1pb = (const float*)d_in[6];
    const float* r1nf = (const float*)d_in[7];
    const float* r1nb = (const float*)d_in[8];
    const float* r0f  = (const float*)d_in[9];
    const float* r0b  = (const float*)d_in[10];
    const float* r2f  = (const float*)d_in[11];
    const float* r2b  = (const float*)d_in[12];
    const float* f01f = (const float*)d_in[13];
    const float* f10b = (const float*)d_in[14];
    const float* f12f = (const float*)d_in[15];
    const float* f21b = (const float*)d_in[16];

    float* ws  = (float*)d_ws;          // 2048 partial sums (8 KB)
    float* out = (float*)d_out;

    flowloss_main<<<NBLK, NTHR, 0, stream>>>(roi, hm0, hm1, hm2,
                                             r1pf, r1pb, r1nf, r1nb,
                                             r0f, r0b, r2f, r2b,
                                             f01f, f10b, f12f, f21b, ws);
    flowloss_reduce<<<1, 32, 0, stream>>>(ws, NBLK, out);
}